// HierarchicalCubeMap_87789131530585
// MI455X (gfx1250) — compile-verified
//
#include <hip/hip_runtime.h>
#include <math.h>

typedef float v2f __attribute__((ext_vector_type(2)));
typedef float v8f __attribute__((ext_vector_type(8)));

#define RES0   128
#define RES1   512
#define KHALF  15          // (31-1)/2
#define NTAP   31
#define KPAD   48          // 46 padded to multiple of 4
#define APAD   49          // A-tile row stride: odd -> conflict-free LDS banks
#define PI_F   3.14159265358979323846f

// ---------------------------------------------------------------------------
// Kernel 1: build banded separable-Gaussian weight matrices.
// W[s][p][j] = g_s[p - j] for 0 <= p-j <= 30 else 0,  p in [0,48), j in [0,16)
// ---------------------------------------------------------------------------
__global__ void build_weights_kernel(float* __restrict__ W) {
    int t = blockIdx.x * blockDim.x + threadIdx.x;
    if (t >= 4 * KPAD * 16) return;
    int j  = t & 15;
    int p  = (t >> 4) % KPAD;
    int si = t / (KPAD * 16);
    const float stds[4] = {8.f, 4.f, 2.f, 1.f};
    float s = stds[si];
    float sum = 0.f;
    #pragma unroll
    for (int k = 0; k < NTAP; ++k) {
        float d = (float)(k - KHALF) / s;
        sum += expf(-0.5f * d * d);
    }
    int k = p - j;
    float val = 0.f;
    if (k >= 0 && k < NTAP) {
        float d = (float)(k - KHALF) / s;
        val = expf(-0.5f * d * d) / sum;
    }
    W[t] = val;
}

// ---------------------------------------------------------------------------
// Kernel 2: horizontal blur pass. One wave = one 16x16 tile.
//   out[y0+i][x0+j] = sum_p in[y0+i][x0-15+p] * W[p][j]
// Stage zero-padded 16x48 window + 48x16 weight band in LDS, then a
// branch-free ds_load -> v_wmma_f32_16x16x4_f32 chain (12 chained WMMAs).
// ---------------------------------------------------------------------------
__global__ void hpass_kernel(const float* __restrict__ in,   // bg_mat0 (6,128,128,3)
                             const float* __restrict__ W,
                             float* __restrict__ tmp) {      // (4,6,3,128,128) planar
    __shared__ float tileA[16][APAD];   // [row i][p], zero-padded halo
    __shared__ float Wl[KPAD][16];      // [p][j]

    int job  = blockIdx.x;
    int tile = job & 63;
    int ch   = (job >> 6) % 3;
    int face = (job / 192) % 6;
    int s    = job / 1152;
    int y0 = (tile >> 3) * 16;
    int x0 = (tile & 7) * 16;

    int lane = threadIdx.x;

    // cooperative staging (branch-free WMMA loop afterwards)
    const float* inFace = in + (size_t)face * RES0 * RES0 * 3 + ch;
    for (int idx = lane; idx < 16 * KPAD; idx += 32) {
        int r = idx / KPAD, p = idx % KPAD;
        int gc = x0 - KHALF + p;
        tileA[r][p] = (gc >= 0 && gc < RES0)
                      ? inFace[((size_t)(y0 + r) * RES0 + gc) * 3] : 0.f;
    }
    const float* Ws = W + s * (KPAD * 16);
    for (int idx = lane; idx < KPAD * 16; idx += 32)
        (&Wl[0][0])[idx] = Ws[idx];
    __syncthreads();

    int half = lane >> 4;        // 0: lanes 0-15, 1: lanes 16-31
    int l    = lane & 15;

    v8f c = {};
    #pragma unroll
    for (int kc = 0; kc < KPAD / 4; ++kc) {
        int kb = kc * 4 + half * 2;          // this lane's K rows: kb, kb+1
        v2f a, b;
        a.x = tileA[l][kb];  a.y = tileA[l][kb + 1];   // A (M=l, K=kb..)
        b.x = Wl[kb][l];     b.y = Wl[kb + 1][l];      // B (K=kb.., N=l)
        c = __builtin_amdgcn_wmma_f32_16x16x4_f32(false, a, false, b,
                                                  (short)0, c, false, false);
    }
    // D: vgpr r holds M = r + half*8, N = l
    float* outBase = tmp + ((size_t)((s * 6 + face) * 3 + ch) * RES0) * RES0;
    #pragma unroll
    for (int r = 0; r < 8; ++r) {
        int m = r + half * 8;
        outBase[(y0 + m) * RES0 + (x0 + l)] = c[r];
    }
}

// ---------------------------------------------------------------------------
// Kernel 3: vertical blur pass.
//   out[y0+i][x0+j] = sum_p W[p][i] * tmp[y0-15+p][x0+j]
// A = banded-weight transpose (read Wl[p][i]), B = staged tmp rows.
// ---------------------------------------------------------------------------
__global__ void vpass_kernel(const float* __restrict__ tmp,
                             const float* __restrict__ W,
                             float* __restrict__ conv) {     // (4,6,3,128,128) planar
    __shared__ float tileB[KPAD][16];   // [p][j], zero-padded halo rows
    __shared__ float Wl[KPAD][16];

    int job  = blockIdx.x;
    int tile = job & 63;
    int ch   = (job >> 6) % 3;
    int face = (job / 192) % 6;
    int s    = job / 1152;
    int y0 = (tile >> 3) * 16;
    int x0 = (tile & 7) * 16;

    int lane = threadIdx.x;

    const float* tmpBase = tmp + ((size_t)((s * 6 + face) * 3 + ch) * RES0) * RES0;
    for (int idx = lane; idx < KPAD * 16; idx += 32) {
        int r = idx / 16, cx = idx % 16;
        int gr = y0 - KHALF + r;
        tileB[r][cx] = (gr >= 0 && gr < RES0)
                       ? tmpBase[(size_t)gr * RES0 + x0 + cx] : 0.f;
    }
    const float* Ws = W + s * (KPAD * 16);
    for (int idx = lane; idx < KPAD * 16; idx += 32)
        (&Wl[0][0])[idx] = Ws[idx];
    __syncthreads();

    int half = lane >> 4;
    int l    = lane & 15;

    v8f c = {};
    #pragma unroll
    for (int kc = 0; kc < KPAD / 4; ++kc) {
        int kb = kc * 4 + half * 2;
        v2f a, b;
        a.x = Wl[kb][l];     a.y = Wl[kb + 1][l];      // A (M=l=i, K=kb..) = W[p][i]
        b.x = tileB[kb][l];  b.y = tileB[kb + 1][l];   // B (K=kb.., N=l)
        c = __builtin_amdgcn_wmma_f32_16x16x4_f32(false, a, false, b,
                                                  (short)0, c, false, false);
    }
    float* outBase = conv + ((size_t)((s * 6 + face) * 3 + ch) * RES0) * RES0;
    #pragma unroll
    for (int r = 0; r < 8; ++r) {
        int m = r + half * 8;
        outBase[(y0 + m) * RES0 + (x0 + l)] = c[r];
    }
}

// ---------------------------------------------------------------------------
// Kernel 4: per-ray cube sampling + mip blending (gather-bound, VALU path).
// ---------------------------------------------------------------------------
struct Bil { int x0, x1, y0, y1; float wx, wy; };

__device__ __forceinline__ Bil make_bil(float u, float v, int R) {
    float fx = u * (float)R - 0.5f;
    float fy = v * (float)R - 0.5f;
    float x0f = floorf(fx), y0f = floorf(fy);
    Bil b;
    b.wx = fx - x0f;
    b.wy = fy - y0f;
    float Rm1 = (float)(R - 1);
    b.x0 = (int)fminf(fmaxf(x0f,        0.f), Rm1);
    b.x1 = (int)fminf(fmaxf(x0f + 1.f,  0.f), Rm1);
    b.y0 = (int)fminf(fmaxf(y0f,        0.f), Rm1);
    b.y1 = (int)fminf(fmaxf(y0f + 1.f,  0.f), Rm1);
    return b;
}

// interleaved (face,y,x,3) texture, accumulate w * bilinear into o[3]
__device__ __forceinline__ void samp3_interleaved(const float* __restrict__ tex,
                                                  int face, int R, const Bil& b,
                                                  float w, float* o) {
    const float* base = tex + (size_t)face * R * R * 3;
    const float* p00 = base + ((size_t)b.y0 * R + b.x0) * 3;
    const float* p10 = base + ((size_t)b.y0 * R + b.x1) * 3;
    const float* p01 = base + ((size_t)b.y1 * R + b.x0) * 3;
    const float* p11 = base + ((size_t)b.y1 * R + b.x1) * 3;
    #pragma unroll
    for (int ch = 0; ch < 3; ++ch) {
        float top = p00[ch] * (1.f - b.wx) + p10[ch] * b.wx;
        float bot = p01[ch] * (1.f - b.wx) + p11[ch] * b.wx;
        o[ch] += w * (top * (1.f - b.wy) + bot * b.wy);
    }
}

// planar (s,face,ch,y,x) conv texture
__device__ __forceinline__ void samp3_planar(const float* __restrict__ plane0,
                                             const Bil& b, float w, float* o) {
    #pragma unroll
    for (int ch = 0; ch < 3; ++ch) {
        const float* pl = plane0 + (size_t)ch * RES0 * RES0;
        float t00 = pl[b.y0 * RES0 + b.x0];
        float t10 = pl[b.y0 * RES0 + b.x1];
        float t01 = pl[b.y1 * RES0 + b.x0];
        float t11 = pl[b.y1 * RES0 + b.x1];
        float top = t00 * (1.f - b.wx) + t10 * b.wx;
        float bot = t01 * (1.f - b.wx) + t11 * b.wx;
        o[ch] += w * (top * (1.f - b.wy) + bot * b.wy);
    }
}

__global__ void sample_kernel(const float* __restrict__ dirs,
                              const float* __restrict__ sa,
                              const float* __restrict__ bg0,
                              const float* __restrict__ bg1,
                              const float* __restrict__ conv,
                              float* __restrict__ out, int B) {
    int i = blockIdx.x * blockDim.x + threadIdx.x;
    if (i >= B) return;

    float x = dirs[(size_t)i * 3 + 0];
    float y = dirs[(size_t)i * 3 + 1];
    float z = dirs[(size_t)i * 3 + 2];
    float ax = fabsf(x), ay = fabsf(y), az = fabsf(z);
    bool is_x = (ax >= ay) && (ax >= az);
    bool is_y = (!is_x) && (ay >= az);

    int face; float sc, tc, ma;
    if (is_x)      { face = (x >= 0.f) ? 0 : 1; ma = ax; sc = (face == 0) ? -z :  z; tc = -y; }
    else if (is_y) { face = (y >= 0.f) ? 2 : 3; ma = ay; sc =  x;                    tc = (face == 2) ? z : -z; }
    else           { face = (z >= 0.f) ? 4 : 5; ma = az; sc = (face == 4) ?  x : -x; tc = -y; }

    float inv = 1.f / fmaxf(ma, 1e-12f);
    float u = 0.5f * (sc * inv + 1.f);
    float v = 0.5f * (tc * inv + 1.f);

    const float saTexel = 4.f * PI_F / (6.f * (float)RES1 * (float)RES1);
    float m = logf(sa[i] / saTexel) * (1.f / (logf(4.f) * 2.f));
    m = fminf(fmaxf(m, 0.f), 3.f);   // MAX_MIP = 3

    float o[3] = {0.f, 0.f, 0.f};

    if (m >= 2.f) {
        // blurred path: blend up to 4 pre-blurred 128^2 mips
        Bil b128 = make_bil(u, v, RES0);
        const float mips[4] = {3.f, 2.5f, 2.f, 1.5f};
        float wsum = 0.f;
        #pragma unroll
        for (int s = 0; s < 4; ++s) {
            float w = 1.f - fminf(fmaxf(fabsf(mips[s] - m), 0.f), 1.f);
            if (w > 0.f) {
                const float* plane0 = conv + (size_t)((s * 6 + face) * 3) * RES0 * RES0;
                samp3_planar(plane0, b128, w, o);
                wsum += w;
            }
        }
        float iw = 1.f / fmaxf(wsum, 1e-8f);
        o[0] *= iw; o[1] *= iw; o[2] *= iw;
    } else {
        Bil b128 = make_bil(u, v, RES0);
        Bil b512 = make_bil(u, v, RES1);
        samp3_interleaved(bg0, face, RES0, b128, 1.0f, o);
        samp3_interleaved(bg1, face, RES1, b512, 0.5f, o);
    }

    out[(size_t)i * 3 + 0] = o[0];
    out[(size_t)i * 3 + 1] = o[1];
    out[(size_t)i * 3 + 2] = o[2];
}

// ---------------------------------------------------------------------------
// Launch: weights -> H pass (WMMA) -> V pass (WMMA) -> ray sampling
// ---------------------------------------------------------------------------
extern "C" void kernel_launch(void* const* d_in, const int* in_sizes, int n_in,
                              void* d_out, int out_size, void* d_ws, size_t ws_size,
                              hipStream_t stream) {
    const float* viewdirs = (const float*)d_in[0];   // (B,3)
    const float* saSample = (const float*)d_in[1];   // (B,1)
    const float* bg_mat0  = (const float*)d_in[2];   // (6,128,128,3)
    const float* bg_mat1  = (const float*)d_in[3];   // (6,512,512,3)
    float*       out      = (float*)d_out;

    int B = in_sizes[0] / 3;

    // workspace layout
    char* ws = (char*)d_ws;
    float* W    = (float*)ws;                                // 4*48*16*4   = 12 KB
    float* tmpH = (float*)(ws + 16384);                      // 4*6*3*128*128*4 = 4.5 MB
    float* conv = (float*)(ws + 16384 + 4u * 6u * 3u * 128u * 128u * 4u);

    // 1) banded Gaussian weight matrices
    {
        int n = 4 * KPAD * 16;
        build_weights_kernel<<<(n + 255) / 256, 256, 0, stream>>>(W);
    }
    // 2) horizontal pass: 4608 tile-jobs, one wave each
    hpass_kernel<<<4 * 6 * 3 * 64, 32, 0, stream>>>(bg_mat0, W, tmpH);
    // 3) vertical pass
    vpass_kernel<<<4 * 6 * 3 * 64, 32, 0, stream>>>(tmpH, W, conv);
    // 4) per-ray gather + blend
    sample_kernel<<<(B + 255) / 256, 256, 0, stream>>>(viewdirs, saSample,
                                                       bg_mat0, bg_mat1, conv,
                                                       out, B);
    (void)n_in; (void)out_size; (void)ws_size;
}